// DecoderRNN_83880711291415
// MI455X (gfx1250) — compile-verified
//
#include <hip/hip_runtime.h>
#include <math.h>

#define HDIM 1024
#define SLEN 2048
#define VOCAB 50257

typedef __attribute__((ext_vector_type(2))) float v2f;
typedef __attribute__((ext_vector_type(8))) float v8f;

__device__ __forceinline__ float sigmoid_f(float x) {
    return 1.0f / (1.0f + expf(-x));
}

// ---------------------------------------------------------------------------
// Generic fp32 GEMV via V_WMMA_F32_16X16X4_F32.
// out[r] = act( dot(M[r, 0:K], vec) + bias[r] )
// Block = 256 threads = 8 waves; each wave computes a 16-row tile.
// K must be a multiple of 8 and <= 2048.
// ---------------------------------------------------------------------------
__global__ __launch_bounds__(256)
void gemv_wmma_f32(const float* __restrict__ M, const float* __restrict__ vec,
                   const float* __restrict__ bias, float* __restrict__ out,
                   int rows, int K, int act)
{
    __shared__ __align__(16) float vlds[2048];
    const int tid = threadIdx.x;
    for (int i = tid; i < K; i += 256) vlds[i] = vec[i];
    __syncthreads();

    const int lane    = tid & 31;
    const int wave    = tid >> 5;
    const int rowBase = (blockIdx.x * 8 + wave) * 16;
    if (rowBase >= rows) return;                 // uniform per-wave exit

    const int lrow  = rowBase + (lane & 15);
    const int row   = (lrow < rows) ? lrow : (rows - 1);   // clamp ragged tile
    const int khalf = (lane >> 4) << 2;                    // 0 (low half) or 4 (high half)
    const bool isN0 = (lane & 15) == 0;                    // lanes 0 and 16 carry N=0

    const float4* Arow = (const float4*)(M + (size_t)row * (size_t)K);
    const float4* Blds = (const float4*)vlds;

    v8f acc = {0.f, 0.f, 0.f, 0.f, 0.f, 0.f, 0.f, 0.f};

    for (int k0 = 0; k0 < K; k0 += 8) {
        const int q = (k0 + khalf) >> 2;
        float4 a = Arow[q];      // 16 bytes of this lane's row
        float4 b = Blds[q];      // LDS broadcast (2 addresses per wave)
        float4 bm;
        bm.x = isN0 ? b.x : 0.0f;
        bm.y = isN0 ? b.y : 0.0f;
        bm.z = isN0 ? b.z : 0.0f;
        bm.w = isN0 ? b.w : 0.0f;

        v2f A0 = {a.x, a.y};     // low lanes: K slots 0,1 = k0,k0+1 ; high: slots 2,3 = k0+4,k0+5
        v2f A1 = {a.z, a.w};     // low: k0+2,k0+3 ; high: k0+6,k0+7
        v2f B0 = {bm.x, bm.y};   // matching K-permutation on B (we own the vector layout)
        v2f B1 = {bm.z, bm.w};

        acc = __builtin_amdgcn_wmma_f32_16x16x4_f32(false, A0, false, B0,
                                                    (short)0, acc, false, false);
        acc = __builtin_amdgcn_wmma_f32_16x16x4_f32(false, A1, false, B1,
                                                    (short)0, acc, false, false);
    }

    // D[:,0]: lane 0 holds rows rowBase+0..7 in acc[0..7]; lane 16 holds rows +8..15
    if ((lane & 15) == 0) {
        const int r0 = rowBase + ((lane >> 4) << 3);
#pragma unroll
        for (int m = 0; m < 8; ++m) {
            const int r = r0 + m;
            if (r < rows) {
                float val = acc[m];
                if (bias) val += bias[r];
                if (act)  val = tanhf(val);
                out[r] = val;
            }
        }
    }
}

// ---------------------------------------------------------------------------
// x = relu(emb_table[id])
// ---------------------------------------------------------------------------
__global__ void embed_relu_kernel(const int* __restrict__ ids,
                                  const float* __restrict__ emb,
                                  float* __restrict__ x)
{
    const int j = blockIdx.x * blockDim.x + threadIdx.x;
    if (j < HDIM) {
        float v = emb[(size_t)ids[0] * HDIM + j];
        x[j] = v > 0.0f ? v : 0.0f;
    }
}

// ---------------------------------------------------------------------------
// GRU gates (PyTorch order r, z, n): h_new = (1-z)*n + z*h
// ---------------------------------------------------------------------------
__global__ void gru_gates_kernel(const float* __restrict__ gi,
                                 const float* __restrict__ gh,
                                 const float* __restrict__ h,
                                 float* __restrict__ hnew_ws,
                                 float* __restrict__ hnew_out)
{
    const int j = blockIdx.x * blockDim.x + threadIdx.x;
    if (j < HDIM) {
        float r = sigmoid_f(gi[j]           + gh[j]);
        float z = sigmoid_f(gi[HDIM + j]    + gh[HDIM + j]);
        float n = tanhf(gi[2 * HDIM + j] + r * gh[2 * HDIM + j]);
        float hn = (1.0f - z) * n + z * h[j];
        hnew_ws[j]  = hn;
        hnew_out[j] = hn;
    }
}

// ---------------------------------------------------------------------------
// attn = softmax(scores) over S=2048; also zeroes ctx accumulator (H floats).
// Single block of 1024 threads.
// ---------------------------------------------------------------------------
__global__ __launch_bounds__(1024)
void attn_softmax_kernel(const float* __restrict__ scores,
                         float* __restrict__ attn_ws,
                         float* __restrict__ attn_out,
                         float* __restrict__ ctx_zero)
{
    __shared__ float red[1024];
    const int t = threadIdx.x;

    float m = -3.4e38f;
    for (int i = t; i < SLEN; i += 1024) m = fmaxf(m, scores[i]);
    red[t] = m; __syncthreads();
    for (int s = 512; s > 0; s >>= 1) {
        if (t < s) red[t] = fmaxf(red[t], red[t + s]);
        __syncthreads();
    }
    const float mx = red[0]; __syncthreads();

    float sum = 0.0f;
    for (int i = t; i < SLEN; i += 1024) sum += expf(scores[i] - mx);
    red[t] = sum; __syncthreads();
    for (int s = 512; s > 0; s >>= 1) {
        if (t < s) red[t] += red[t + s];
        __syncthreads();
    }
    const float inv = 1.0f / red[0];

    for (int i = t; i < SLEN; i += 1024) {
        float a = expf(scores[i] - mx) * inv;
        attn_ws[i]  = a;
        attn_out[i] = a;
    }
    if (t < HDIM) ctx_zero[t] = 0.0f;   // ctx accumulator for next kernel
}

// ---------------------------------------------------------------------------
// ctx[j] += sum_i attn[i] * enc[i][j] over this block's S-chunk.
// grid = (HDIM/256, 16), block = 256.
// ---------------------------------------------------------------------------
__global__ void ctx_kernel(const float* __restrict__ enc,
                           const float* __restrict__ attn,
                           float* __restrict__ ctx)
{
    const int j  = blockIdx.x * 256 + threadIdx.x;
    const int i0 = blockIdx.y * (SLEN / 16);
    float acc = 0.0f;
    for (int i = i0; i < i0 + SLEN / 16; ++i)
        acc += attn[i] * enc[(size_t)i * HDIM + j];
    atomicAdd(&ctx[j], acc);
}

// ---------------------------------------------------------------------------
// log_probs = log_softmax(logits) over V=50257. Single block of 1024 threads.
// ---------------------------------------------------------------------------
__global__ __launch_bounds__(1024)
void log_softmax_kernel(const float* __restrict__ logits,
                        float* __restrict__ out)
{
    __shared__ float red[1024];
    const int t = threadIdx.x;

    float m = -3.4e38f;
    for (int i = t; i < VOCAB; i += 1024) m = fmaxf(m, logits[i]);
    red[t] = m; __syncthreads();
    for (int s = 512; s > 0; s >>= 1) {
        if (t < s) red[t] = fmaxf(red[t], red[t + s]);
        __syncthreads();
    }
    const float mx = red[0]; __syncthreads();

    float sum = 0.0f;
    for (int i = t; i < VOCAB; i += 1024) sum += expf(logits[i] - mx);
    red[t] = sum; __syncthreads();
    for (int s = 512; s > 0; s >>= 1) {
        if (t < s) red[t] += red[t + s];
        __syncthreads();
    }
    const float lse = mx + logf(red[0]);

    for (int i = t; i < VOCAB; i += 1024) out[i] = logits[i] - lse;
}

// ---------------------------------------------------------------------------
// Host-side launch sequence (all on `stream`, scratch in d_ws)
// ---------------------------------------------------------------------------
extern "C" void kernel_launch(void* const* d_in, const int* in_sizes, int n_in,
                              void* d_out, int out_size, void* d_ws, size_t ws_size,
                              hipStream_t stream)
{
    (void)in_sizes; (void)n_in; (void)out_size; (void)ws_size;

    const int*   ids    = (const int*)  d_in[0];
    const float* hidden = (const float*)d_in[1];   // (1,1,H)
    const float* enc    = (const float*)d_in[2];   // (S,H)
    const float* emb    = (const float*)d_in[3];   // (V,H)
    const float* W_ih   = (const float*)d_in[4];   // (3H,H)
    const float* W_hh   = (const float*)d_in[5];   // (3H,H)
    const float* b_ih   = (const float*)d_in[6];
    const float* b_hh   = (const float*)d_in[7];
    const float* W_comb = (const float*)d_in[8];   // (H,2H)
    const float* b_comb = (const float*)d_in[9];
    const float* W_out  = (const float*)d_in[10];  // (V,H)
    const float* b_out  = (const float*)d_in[11];

    float* out_logp = (float*)d_out;               // V
    float* out_h    = out_logp + VOCAB;            // H
    float* out_attn = out_h + HDIM;                // S

    // workspace layout (floats)
    float* ws      = (float*)d_ws;
    float* x       = ws;                 // 1024
    float* gi      = x + HDIM;           // 3072
    float* gh      = gi + 3 * HDIM;      // 3072
    float* catbuf  = gh + 3 * HDIM;      // 2048: [ctx | h_new]
    float* hnew    = catbuf + HDIM;
    float* scores  = catbuf + 2 * HDIM;  // 2048 (reused as attn)
    float* comb    = scores + SLEN;      // 1024
    float* logits  = comb + HDIM;        // 50257

    // 1. x = relu(emb[id])
    embed_relu_kernel<<<(HDIM + 255) / 256, 256, 0, stream>>>(ids, emb, x);

    // 2-3. GRU input/hidden GEMVs (3072 x 1024)
    gemv_wmma_f32<<<(3 * HDIM + 127) / 128, 256, 0, stream>>>(W_ih, x,      b_ih, gi, 3 * HDIM, HDIM, 0);
    gemv_wmma_f32<<<(3 * HDIM + 127) / 128, 256, 0, stream>>>(W_hh, hidden, b_hh, gh, 3 * HDIM, HDIM, 0);

    // 4. gates -> h_new (to ws and to output slot)
    gru_gates_kernel<<<(HDIM + 255) / 256, 256, 0, stream>>>(gi, gh, hidden, hnew, out_h);

    // 5. attention scores = enc @ h_new   (2048 x 1024)
    gemv_wmma_f32<<<(SLEN + 127) / 128, 256, 0, stream>>>(enc, hnew, nullptr, scores, SLEN, HDIM, 0);

    // 6. softmax -> attn (ws + output), zero ctx accumulator
    attn_softmax_kernel<<<1, 1024, 0, stream>>>(scores, scores, out_attn, catbuf);

    // 7. ctx = attn @ enc  (column reduction with per-block partials)
    ctx_kernel<<<dim3(HDIM / 256, 16), 256, 0, stream>>>(enc, scores, catbuf);

    // 8. comb = tanh(W_comb @ [ctx; h_new] + b_comb)   (1024 x 2048)
    gemv_wmma_f32<<<(HDIM + 127) / 128, 256, 0, stream>>>(W_comb, catbuf, b_comb, comb, HDIM, 2 * HDIM, 1);

    // 9. logits = W_out @ comb + b_out   (50257 x 1024 — dominant, ~206 MB)
    gemv_wmma_f32<<<(VOCAB + 127) / 128, 256, 0, stream>>>(W_out, comb, b_out, logits, VOCAB, HDIM, 0);

    // 10. log_softmax -> output
    log_softmax_kernel<<<1, 1024, 0, stream>>>(logits, out_logp);
}